// AllGraph_35296041238627
// MI455X (gfx1250) — compile-verified
//
#include <hip/hip_runtime.h>
#include <hip/hip_bf16.h>

typedef float v2f __attribute__((ext_vector_type(2)));
typedef float v8f __attribute__((ext_vector_type(8)));

#define LRELU 0.2f
#define LN_EPS 1e-5f
#define SM_EPS 1e-16f

#define KC 32            // K-chunk staged in LDS
#define BS_STRIDE 68     // 64 cols + 4 floats pad (break 2-way bank conflict)

// ---------------------------------------------------------------------------
// C[n x 512] = A[n x 512] * B[512 x 512], fp32 via V_WMMA_F32_16X16X4_F32.
// Block tile: 128 rows x 64 cols (8 waves x 16 rows, all sharing one column
// strip). B chunks (32x64 fp32) are staged into LDS with
// GLOBAL_LOAD_ASYNC_TO_LDS_B128 (ASYNCcnt), double-buffered so the DMA of
// chunk i+1 overlaps WMMA on chunk i. A rows are per-wave global b64 loads
// (L2-resident, no intra-block reuse to exploit).
// Fragment layouts per CDNA5 ISA 7.12.2:
//   A 16x4 : lanes 0-15 M=0..15, VGPR0={K0|K2}, VGPR1={K1|K3}  (| split at lane16)
//   B 4x16 : VGPR0={K0|K2}, VGPR1={K1|K3}, N striped across lanes
//   C 16x16: VGPR r = rows {r | r+8}, N striped across lanes
// ---------------------------------------------------------------------------
__global__ void gemm512_wmma_f32(const float* __restrict__ A,
                                 const float* __restrict__ B,
                                 float* __restrict__ C, int n) {
  const int K = 512, NB = 512;
  __shared__ float Bs[2][KC * BS_STRIDE];

  int t    = threadIdx.x;
  int wid  = t >> 5;
  int lane = t & 31;
  int n0   = (blockIdx.x & 7) * 64;
  int m0   = (blockIdx.x >> 3) * 128 + wid * 16;
  int half = lane >> 4;
  int l    = lane & 15;

  // async-copy coordinates: each thread moves two 16B chunks per K-chunk
  int r0 = t >> 4;   // row 0..15 (and r0+16)
  int o0 = t & 15;   // 16B chunk within the 256B row

  const float* Arow = A + (size_t)(m0 + l) * K;

  v8f acc0 = {}, acc1 = {}, acc2 = {}, acc3 = {};

  // prologue: async-stage chunk 0 into buffer 0
  {
    unsigned long long g0 = (unsigned long long)(const void*)(B + (size_t)r0 * NB + n0 + o0 * 4);
    unsigned long long g1 = (unsigned long long)(const void*)(B + (size_t)(r0 + 16) * NB + n0 + o0 * 4);
    unsigned l0 = (unsigned)(uintptr_t)&Bs[0][r0 * BS_STRIDE + o0 * 4];
    unsigned l1 = (unsigned)(uintptr_t)&Bs[0][(r0 + 16) * BS_STRIDE + o0 * 4];
    asm volatile("global_load_async_to_lds_b128 %0, %1, off" :: "v"(l0), "v"(g0) : "memory");
    asm volatile("global_load_async_to_lds_b128 %0, %1, off" :: "v"(l1), "v"(g1) : "memory");
  }

  const int NCH = K / KC;  // 16
  for (int ch = 0; ch < NCH; ++ch) {
    int cur = ch & 1;
    if (ch + 1 < NCH) {
      int nxt = cur ^ 1;
      int kc  = (ch + 1) * KC;
      unsigned long long g0 = (unsigned long long)(const void*)(B + (size_t)(kc + r0) * NB + n0 + o0 * 4);
      unsigned long long g1 = (unsigned long long)(const void*)(B + (size_t)(kc + r0 + 16) * NB + n0 + o0 * 4);
      unsigned l0 = (unsigned)(uintptr_t)&Bs[nxt][r0 * BS_STRIDE + o0 * 4];
      unsigned l1 = (unsigned)(uintptr_t)&Bs[nxt][(r0 + 16) * BS_STRIDE + o0 * 4];
      asm volatile("global_load_async_to_lds_b128 %0, %1, off" :: "v"(l0), "v"(g0) : "memory");
      asm volatile("global_load_async_to_lds_b128 %0, %1, off" :: "v"(l1), "v"(g1) : "memory");
      asm volatile("s_wait_asynccnt 0x2" ::: "memory");   // our older chunk is done (in-order)
    } else {
      asm volatile("s_wait_asynccnt 0x0" ::: "memory");
    }
    __syncthreads();

    const float* Ab = Arow + ch * KC;
    const float* Bb = &Bs[cur][0];
#pragma unroll
    for (int kk = 0; kk < KC; kk += 4) {
      v2f a;
      a.x = Ab[kk + 2 * half];
      a.y = Ab[kk + 2 * half + 1];
      const float* bp = Bb + (kk + 2 * half) * BS_STRIDE + l;
      v2f b0, b1, b2, b3;
      b0.x = bp[0];   b0.y = bp[BS_STRIDE];
      b1.x = bp[16];  b1.y = bp[BS_STRIDE + 16];
      b2.x = bp[32];  b2.y = bp[BS_STRIDE + 32];
      b3.x = bp[48];  b3.y = bp[BS_STRIDE + 48];
      acc0 = __builtin_amdgcn_wmma_f32_16x16x4_f32(false, a, false, b0, (short)0, acc0, false, false);
      acc1 = __builtin_amdgcn_wmma_f32_16x16x4_f32(false, a, false, b1, (short)0, acc1, false, false);
      acc2 = __builtin_amdgcn_wmma_f32_16x16x4_f32(false, a, false, b2, (short)0, acc2, false, false);
      acc3 = __builtin_amdgcn_wmma_f32_16x16x4_f32(false, a, false, b3, (short)0, acc3, false, false);
    }
    __syncthreads();
  }

  float* Crow = C + (size_t)(m0 + 8 * half) * NB + n0 + l;
#pragma unroll
  for (int r = 0; r < 8; ++r) {
    Crow[(size_t)r * NB +  0] = acc0[r];
    Crow[(size_t)r * NB + 16] = acc1[r];
    Crow[(size_t)r * NB + 32] = acc2[r];
    Crow[(size_t)r * NB + 48] = acc3[r];
  }
}

// al_s[n,H] = sum_c xt[n,h,c]*a_src[h,c]; same for al_d. One thread per (node,head).
__global__ void attn_logits(const float* __restrict__ xt,
                            const float* __restrict__ a_src,
                            const float* __restrict__ a_dst,
                            float* __restrict__ al_s, float* __restrict__ al_d,
                            int n, int H, int C) {
  int idx = blockIdx.x * blockDim.x + threadIdx.x;
  if (idx >= n * H) return;
  int node = idx / H, h = idx - node * H;
  const float* row = xt + (size_t)node * H * C + (size_t)h * C;
  float ss = 0.f, sd = 0.f;
  for (int c = 0; c < C; ++c) {
    float v = row[c];
    ss += v * a_src[h * C + c];
    sd += v * a_dst[h * C + c];
  }
  al_s[idx] = ss;
  al_d[idx] = sd;
}

// monotone f32 <-> u32 key so segment-max can use native u32 atomicMax
__device__ __forceinline__ unsigned f2key(float f) {
  unsigned b = __float_as_uint(f);
  return (b & 0x80000000u) ? ~b : (b | 0x80000000u);
}
__device__ __forceinline__ float key2f(unsigned k) {
  return __uint_as_float((k & 0x80000000u) ? (k & 0x7fffffffu) : ~k);
}

// e = leaky_relu(al_s[src]+al_d[dst]); segment max over dst
__global__ void edge_logit_max(const int* __restrict__ src, const int* __restrict__ dst,
                               const float* __restrict__ al_s, const float* __restrict__ al_d,
                               float* __restrict__ ebuf, unsigned* __restrict__ mkey,
                               int E, int H) {
  int idx = blockIdx.x * blockDim.x + threadIdx.x;
  if (idx >= E * H) return;
  int e = idx / H, h = idx - e * H;
  int s = src[e], d = dst[e];
  float v = al_s[s * H + h] + al_d[d * H + h];
  v = v > 0.f ? v : LRELU * v;
  ebuf[idx] = v;
  atomicMax(&mkey[d * H + h], f2key(v));
}

// ex = exp(e - m[dst]); denom[dst] += ex  (ex overwrites ebuf)
__global__ void edge_exp_sum(const int* __restrict__ dst,
                             float* __restrict__ ebuf,
                             const unsigned* __restrict__ mkey,
                             float* __restrict__ denom, int E, int H) {
  int idx = blockIdx.x * blockDim.x + threadIdx.x;
  if (idx >= E * H) return;
  int e = idx / H, h = idx - e * H;
  int d = dst[e];
  float m  = key2f(mkey[d * H + h]);
  float ex = expf(ebuf[idx] - m);
  ebuf[idx] = ex;
  atomicAdd(&denom[d * H + h], ex);
}

// out[dst] += xt[src] * alpha ; one thread per (edge, 4-channel group)
__global__ void edge_scatter(const int* __restrict__ src, const int* __restrict__ dst,
                             const float* __restrict__ xt, const float* __restrict__ ebuf,
                             const float* __restrict__ denom, float* __restrict__ out,
                             int E, int H, int C) {
  const int D = H * C;
  const int groups = D >> 2;
  long long idx = (long long)blockIdx.x * blockDim.x + threadIdx.x;
  if (idx >= (long long)E * groups) return;
  int e = (int)(idx / groups);
  int g = (int)(idx - (long long)e * groups);
  int dch = g << 2;
  int h = dch / C;
  int s = src[e], d = dst[e];
  float alpha = ebuf[e * H + h] / (denom[d * H + h] + SM_EPS);
  const float4 v = *(const float4*)(xt + (size_t)s * D + dch);
  float* o = out + (size_t)d * D + dch;
  atomicAdd(o + 0, v.x * alpha);
  atomicAdd(o + 1, v.y * alpha);
  atomicAdd(o + 2, v.z * alpha);
  atomicAdd(o + 3, v.w * alpha);
}

// out = LayerNorm(gat_out + bias) * g + beta ; one 256-thread block per node, D=512
__global__ void bias_layernorm512(const float* __restrict__ gin,
                                  const float* __restrict__ b,
                                  const float* __restrict__ g,
                                  const float* __restrict__ beta,
                                  float* __restrict__ out) {
  const int D = 512;
  int node = blockIdx.x;
  int t = threadIdx.x;
  const float* row = gin + (size_t)node * D;
  float v0 = row[t] + b[t];
  float v1 = row[t + 256] + b[t + 256];
  float s = v0 + v1;
  float q = v0 * v0 + v1 * v1;
#pragma unroll
  for (int off = 16; off > 0; off >>= 1) {      // wave32 reduction
    s += __shfl_down(s, off, 32);
    q += __shfl_down(q, off, 32);
  }
  __shared__ float ls[8], lq[8];
  __shared__ float mu_s, rs_s;
  int wid = t >> 5, lane = t & 31;
  if (lane == 0) { ls[wid] = s; lq[wid] = q; }
  __syncthreads();
  if (t == 0) {
    float ts = 0.f, tq = 0.f;
    for (int i = 0; i < 8; ++i) { ts += ls[i]; tq += lq[i]; }
    float mu  = ts / D;
    float var = tq / D - mu * mu;
    mu_s = mu;
    rs_s = rsqrtf(var + LN_EPS);
  }
  __syncthreads();
  float mu = mu_s, rs = rs_s;
  out[(size_t)node * D + t]       = (v0 - mu) * rs * g[t]       + beta[t];
  out[(size_t)node * D + t + 256] = (v1 - mu) * rs * g[t + 256] + beta[t + 256];
}

// ---------------------------------------------------------------------------
// Host-side: one full GATConv + LayerNorm
// ---------------------------------------------------------------------------
static void run_gat(const float* x, int n,
                    const int* src, const int* dst, int E,
                    const float* W, const float* as, const float* ad,
                    const float* b, const float* g, const float* beta,
                    int H, int C, float* out,
                    float* xt, float* al_s, float* al_d, float* ebuf,
                    unsigned* mkey, float* denom, float* gout,
                    hipStream_t stream) {
  const int D = H * C;  // always 512
  gemm512_wmma_f32<<<(n / 128) * 8, 256, 0, stream>>>(x, W, xt, n);
  attn_logits<<<(n * H + 255) / 256, 256, 0, stream>>>(xt, as, ad, al_s, al_d, n, H, C);
  hipMemsetAsync(mkey,  0, (size_t)n * H * sizeof(unsigned), stream);
  hipMemsetAsync(denom, 0, (size_t)n * H * sizeof(float), stream);
  hipMemsetAsync(gout,  0, (size_t)n * D * sizeof(float), stream);
  edge_logit_max<<<(E * H + 255) / 256, 256, 0, stream>>>(src, dst, al_s, al_d, ebuf, mkey, E, H);
  edge_exp_sum<<<(E * H + 255) / 256, 256, 0, stream>>>(dst, ebuf, mkey, denom, E, H);
  long long nthr = (long long)E * (D >> 2);
  edge_scatter<<<(unsigned)((nthr + 255) / 256), 256, 0, stream>>>(src, dst, xt, ebuf, denom, gout, E, H, C);
  bias_layernorm512<<<n, 256, 0, stream>>>(gout, b, g, beta, out);
}

extern "C" void kernel_launch(void* const* d_in, const int* in_sizes, int n_in,
                              void* d_out, int out_size, void* d_ws, size_t ws_size,
                              hipStream_t stream) {
  const int N = 8192, E = 131072, E2 = 65536, GRP = 2048, D = 512;

  const float* x   = (const float*)d_in[0];
  const int*   ei  = (const int*)d_in[1];
  const int*   eim = (const int*)d_in[2];
  const float* W1  = (const float*)d_in[4];
  const float* as1 = (const float*)d_in[5];
  const float* ad1 = (const float*)d_in[6];
  const float* b1  = (const float*)d_in[7];
  const float* g1  = (const float*)d_in[8];
  const float* be1 = (const float*)d_in[9];
  const float* W2  = (const float*)d_in[10];
  const float* as2 = (const float*)d_in[11];
  const float* ad2 = (const float*)d_in[12];
  const float* b2  = (const float*)d_in[13];
  const float* g2  = (const float*)d_in[14];
  const float* be2 = (const float*)d_in[15];

  const int* s  = ei;
  const int* d  = ei + E;
  const int* sm = eim;
  const int* dm = eim + E2;

  float* ws = (float*)d_ws;
  size_t off = 0;
  float* bufA  = ws + off; off += (size_t)N * D;   // 4M floats
  float* bufB  = ws + off; off += (size_t)N * D;   // 4M
  float* xt    = ws + off; off += (size_t)N * D;   // 4M
  float* gout  = ws + off; off += (size_t)N * D;   // 4M
  float* al_s  = ws + off; off += (size_t)N * 8;
  float* al_d  = ws + off; off += (size_t)N * 8;
  float* ebuf  = ws + off; off += (size_t)E * 8;   // 1M
  unsigned* mkey = (unsigned*)(ws + off); off += (size_t)N * 8;
  float* denom = ws + off; off += (size_t)N * 8;

  const size_t WD = (size_t)D * D;   // 512*512 weight block per stage
  const size_t AD = 512;             // a_src/a_dst/bias/gamma/beta block per stage

  // ----- stage 1 (H=8, C=64) -----
  run_gat(x, N, s, d, E, W1, as1, ad1, b1, g1, be1, 8, 64, bufA,
          xt, al_s, al_d, ebuf, mkey, denom, gout, stream);                     // up1
  run_gat(bufA + (size_t)GRP * D, N - GRP, sm, dm, E2,
          W1 + WD, as1 + AD, ad1 + AD, b1 + AD, g1 + AD, be1 + AD, 8, 64,
          bufA + (size_t)GRP * D,
          xt, al_s, al_d, ebuf, mkey, denom, gout, stream);                     // ag1 in place
  run_gat(bufA, N, s, d, E,
          W1 + 2 * WD, as1 + 2 * AD, ad1 + 2 * AD, b1 + 2 * AD, g1 + 2 * AD, be1 + 2 * AD,
          8, 64, bufB,
          xt, al_s, al_d, ebuf, mkey, denom, gout, stream);                     // h

  // ----- stage 2 (H=1, C=512) -----
  run_gat(bufB, N, s, d, E, W2, as2, ad2, b2, g2, be2, 1, 512, bufA,
          xt, al_s, al_d, ebuf, mkey, denom, gout, stream);                     // up2
  run_gat(bufA + (size_t)GRP * D, N - GRP, sm, dm, E2,
          W2 + WD, as2 + AD, ad2 + AD, b2 + AD, g2 + AD, be2 + AD, 1, 512,
          bufA + (size_t)GRP * D,
          xt, al_s, al_d, ebuf, mkey, denom, gout, stream);                     // ag2 in place
  run_gat(bufA, N, s, d, E,
          W2 + 2 * WD, as2 + 2 * AD, ad2 + 2 * AD, b2 + 2 * AD, g2 + 2 * AD, be2 + 2 * AD,
          1, 512, (float*)d_out,
          xt, al_s, al_d, ebuf, mkey, denom, gout, stream);                     // out
}